// PAAC_51488067944789
// MI455X (gfx1250) — compile-verified
//
#include <hip/hip_runtime.h>
#include <hip/hip_bf16.h>

// Problem constants (match reference)
#define N_USERS 100000
#define N_ITEMS 50000
#define N_NODES 150000           // multiple of 16 -> full-EXEC WMMA waves
#define EMB     64
#define LAYERS  3
#define NNZ_E   4000000
#define EPS_    0.1f

typedef __attribute__((ext_vector_type(2))) float v2f;
typedef __attribute__((ext_vector_type(8))) float v8f;

// ---------------- utility kernels ----------------

__global__ void PAAC_zero_f4(float* __restrict__ p, int n4) {
    int i = blockIdx.x * blockDim.x + threadIdx.x;
    if (i < n4) ((float4*)p)[i] = make_float4(0.f, 0.f, 0.f, 0.f);
}

__global__ void PAAC_init_concat(const float* __restrict__ uw,
                                 const float* __restrict__ iw,
                                 float* __restrict__ x) {
    int i = blockIdx.x * blockDim.x + threadIdx.x;          // over N_NODES*EMB/4
    const int nU4 = N_USERS * EMB / 4;
    const int nT4 = N_NODES * EMB / 4;
    if (i >= nT4) return;
    float4 v = (i < nU4) ? ((const float4*)uw)[i] : ((const float4*)iw)[i - nU4];
    ((float4*)x)[i] = v;
}

// ---------------- COO SpMM: y[r] += v * x[c] ----------------
// One wave per edge; lane handles 2 consecutive dims (256B coalesced per edge).
__global__ void PAAC_spmm_coo(const float* __restrict__ x,
                              const float* __restrict__ vals,
                              const int*   __restrict__ rows,
                              const int*   __restrict__ cols,
                              float* __restrict__ y) {
    const int lane = threadIdx.x & 31;
    const int e    = (blockIdx.x * blockDim.x + threadIdx.x) >> 5;   // wave id == edge
    if (e >= NNZ_E) return;

    // gfx1250 global_prefetch_b8 on the streaming edge index/value arrays
    const int pf = e + 512;
    if (pf < NNZ_E) {
        __builtin_prefetch(rows + pf, 0, 0);
        __builtin_prefetch(cols + pf, 0, 0);
        __builtin_prefetch(vals + pf, 0, 0);
    }

    const int   r = rows[e];
    const int   c = cols[e];
    const float v = vals[e];

    const float2 xv = ((const float2*)(x + (size_t)c * EMB))[lane];
    float* yp = y + (size_t)r * EMB + 2 * lane;
    atomicAdd(yp,     v * xv.x);
    atomicAdd(yp + 1, v * xv.y);
}

// ---------------- combine: ego = y + sign(y)*normalize(noise)*EPS ; acc += ego/3
// One wave handles 16 rows. Row L2-norms^2 come from the diagonal of
// noise_tile * noise_tile^T computed with chained V_WMMA_F32_16X16X4_F32.
// For the 16x16x4 f32 layouts, the B (4x16) fragment transposed equals the A
// (16x4) fragment per-lane, so we pass the same registers for A and B.
// All 16 fragments are preloaded before the WMMA chain so the compiler can
// clause the loads instead of stalling load->wait->wmma per slice.
__global__ void PAAC_combine_wmma(float* __restrict__ xio,          // in: spmm result, out: ego
                                  const float* __restrict__ noise_l,
                                  float* __restrict__ acc) {
    const int lane    = threadIdx.x & 31;
    const int wave    = (blockIdx.x * blockDim.x + threadIdx.x) >> 5;
    const int rowbase = wave * 16;
    if (rowbase >= N_NODES) return;                 // wave-uniform; EXEC stays all-1s inside

    const int rloc = lane & 15;                     // A fragment: M = lane%16
    const int koff = (lane >> 4) << 1;              // K offset 0 (lanes 0-15) or 2 (lanes 16-31)
    const float* nrow = noise_l + (size_t)(rowbase + rloc) * EMB;

    // Preload all 16 K-slice fragments (independent loads -> MLP)
    v2f frag[16];
#pragma unroll
    for (int s = 0; s < 16; ++s) {
        float2 t = *(const float2*)(nrow + 4 * s + koff);
        frag[s].x = t.x;                            // VGPR0 = K(koff + 4s)
        frag[s].y = t.y;                            // VGPR1 = K(koff + 4s + 1)
    }

    v8f cacc = {};                                  // 16x16 Gram accumulator
#pragma unroll
    for (int s = 0; s < 16; ++s) {
        cacc = __builtin_amdgcn_wmma_f32_16x16x4_f32(
            /*neg_a=*/false, frag[s], /*neg_b=*/false, frag[s],
            /*c_mod=*/(short)0, cacc, /*reuse_a=*/false, /*reuse_b=*/false);
    }

    // Diagonal element for row m sits in C VGPR (m%8): lanes 0-15 hold M=vgpr,
    // lanes 16-31 hold M=vgpr+8, with N = lane%16. Diagonal => lane m (m<8) or m+16.
#pragma unroll
    for (int m = 0; m < 16; ++m) {
        float n2    = __shfl(cacc[m & 7], (m < 8) ? m : (m + 16), 32);
        float scale = EPS_ / fmaxf(sqrtf(n2), 1e-12f);

        const size_t idx = (size_t)(rowbase + m) * EMB + 2 * lane;
        float2 y2 = *(float2*)(xio + idx);
        float2 nz = *(const float2*)(noise_l + idx);
        float sx = (y2.x > 0.f) ? 1.f : ((y2.x < 0.f) ? -1.f : 0.f);
        float sy = (y2.y > 0.f) ? 1.f : ((y2.y < 0.f) ? -1.f : 0.f);
        float2 e;
        e.x = y2.x + sx * nz.x * scale;
        e.y = y2.y + sy * nz.y * scale;
        *(float2*)(xio + idx) = e;                  // becomes next layer's input

        float2 a2 = *(float2*)(acc + idx);          // exclusive per-lane element: no atomics
        a2.x += e.x * (1.0f / LAYERS);
        a2.y += e.y * (1.0f / LAYERS);
        *(float2*)(acc + idx) = a2;
    }
}

// ---------------- launch ----------------

extern "C" void kernel_launch(void* const* d_in, const int* in_sizes, int n_in,
                              void* d_out, int out_size, void* d_ws, size_t ws_size,
                              hipStream_t stream) {
    const float* user_w = (const float*)d_in[0];
    const float* item_w = (const float*)d_in[1];
    const float* vals   = (const float*)d_in[2];
    const float* noise  = (const float*)d_in[3];   // [LAYERS, N_NODES, EMB]
    const int*   rows   = (const int*)d_in[4];
    const int*   cols   = (const int*)d_in[5];
    float*       out    = (float*)d_out;           // mean embedding, [N_NODES, EMB]

    float* xa = (float*)d_ws;                      // ping
    float* xb = xa + (size_t)N_NODES * EMB;        // pong (ws needs 2 * 38.4 MB)

    const int NT  = N_NODES * EMB;                 // 9,600,000
    const int NT4 = NT / 4;

    PAAC_zero_f4   <<<(NT4 + 255) / 256, 256, 0, stream>>>(out, NT4);
    PAAC_init_concat<<<(NT4 + 255) / 256, 256, 0, stream>>>(user_w, item_w, xa);

    const int spmmBlocks = (NNZ_E * 32 + 255) / 256;          // 1 wave per edge
    const int combBlocks = ((N_NODES / 16) * 32 + 255) / 256; // 1 wave per 16 rows

    for (int l = 0; l < LAYERS; ++l) {
        PAAC_zero_f4     <<<(NT4 + 255) / 256, 256, 0, stream>>>(xb, NT4);
        PAAC_spmm_coo    <<<spmmBlocks, 256, 0, stream>>>(xa, vals, rows, cols, xb);
        PAAC_combine_wmma<<<combBlocks, 256, 0, stream>>>(xb, noise + (size_t)l * NT, out);
        float* t = xa; xa = xb; xb = t;
    }
}